// EmbeddingToProbability_75642964017927
// MI455X (gfx1250) — compile-verified
//
#include <hip/hip_runtime.h>
#include <math.h>

// EmbeddingToProbability for MI455X (gfx1250, wave32).
// prob[n, v] = exp( e_sq[v] + cross[n,v] + c_sq[n] )
//   inv_k   = 1 / (-2*(sigma_k+eps)^2)            (negative)
//   e_sq[v] = sum_k emb[k,v]^2 * inv_k
//   cross   = sum_k emb[k,v] * (-2 * c[n,k] * inv_k)   <-- K=3 GEMM -> WMMA 16x16x4 f32
//   c_sq[n] = sum_k c[n,k]^2 * inv_k
//
// Wave tile: M = 16 centroids (x4 tiles = 64), N = 16 consecutive voxels, K = 4 (3 + zero pad).
// D VGPR j: lanes 0-15 -> (M=j, N=lane), lanes 16-31 -> (M=j+8, N=lane-16),
// so each scalar store writes two full 64B cachelines (16 consecutive voxels / n-plane).

typedef float v2f __attribute__((ext_vector_type(2)));
typedef float v8f __attribute__((ext_vector_type(8)));

#define EPS 1e-16f
#define NCENT 64

__global__ void __launch_bounds__(256)
emb2prob_wmma_kernel(const float* __restrict__ emb,   // [3, V]
                     const float* __restrict__ cent,  // [64, 3]
                     const float* __restrict__ sigma, // [3]
                     float* __restrict__ out,         // [64, V]
                     long V)
{
    __shared__ float sA0[NCENT];   // -2*inv0*c[n][0]
    __shared__ float sA1[NCENT];   // -2*inv1*c[n][1]
    __shared__ float sA2[NCENT];   // -2*inv2*c[n][2]
    __shared__ float sCsq[NCENT];  // sum_k c[n][k]^2 * inv_k

    // Per-thread scalar prep (sigma is 3 floats, hits cache everywhere).
    float s0 = sigma[0] + EPS, s1 = sigma[1] + EPS, s2 = sigma[2] + EPS;
    float inv0 = 1.0f / (-2.0f * s0 * s0);
    float inv1 = 1.0f / (-2.0f * s1 * s1);
    float inv2 = 1.0f / (-2.0f * s2 * s2);

    int t = threadIdx.x;
    if (t < NCENT) {
        float c0 = cent[t * 3 + 0];
        float c1 = cent[t * 3 + 1];
        float c2 = cent[t * 3 + 2];
        sA0[t]  = -2.0f * inv0 * c0;
        sA1[t]  = -2.0f * inv1 * c1;
        sA2[t]  = -2.0f * inv2 * c2;
        sCsq[t] = c0 * c0 * inv0 + c1 * c1 * inv1 + c2 * c2 * inv2;
    }
    __syncthreads();

    const int wave = t >> 5;        // 0..7
    const int lane = t & 31;
    const int hi   = lane >> 4;     // 0: K=0,1 half   1: K=2,pad half
    const int l16  = lane & 15;

    const long base = ((long)blockIdx.x * 8 + wave) * 16;
    if (base + 16 > V) return;      // wave-uniform: EXEC stays all-ones

    const long vox = base + l16;

    // Embedding (channel-major planes). All three channels per lane:
    // halves load identical 64B lines -> broadcast from cache.
    float e0 = emb[vox];
    float e1 = emb[V + vox];
    float e2 = emb[2 * V + vox];

    // B matrix 4x16 (KxN): VGPR0 = rows K=0 (lanes 0-15) / K=2 (lanes 16-31),
    //                      VGPR1 = rows K=1            / K=3 (zero pad).
    v2f b;
    b.x = hi ? e2 : e0;
    b.y = hi ? 0.0f : e1;

    // Per-lane (per-column) e_sq bias.
    const float esq = e0 * e0 * inv0 + e1 * e1 * inv1 + e2 * e2 * inv2;

#pragma unroll
    for (int mt = 0; mt < 4; ++mt) {
        const int nA = mt * 16 + l16;

        // A matrix 16x4 (MxK), M = centroid index, scale folded in.
        v2f a;
        a.x = hi ? sA2[nA] : sA0[nA];
        a.y = hi ? 0.0f : sA1[nA];

        // C = c_sq[row] + e_sq[col]
        v8f c;
#pragma unroll
        for (int j = 0; j < 8; ++j)
            c[j] = sCsq[mt * 16 + j + hi * 8] + esq;

        // D = A x B + C   (v_wmma_f32_16x16x4_f32)
        v8f d = __builtin_amdgcn_wmma_f32_16x16x4_f32(
            /*neg_a=*/false, a, /*neg_b=*/false, b,
            /*c_mod=*/(short)0, c, /*reuse_a=*/false, /*reuse_b=*/false);

        // Each store: lanes 0-15 fill n-plane (mt*16+j), lanes 16-31 fill (mt*16+j+8),
        // 16 consecutive voxels each -> 2 full cachelines per instruction.
#pragma unroll
        for (int j = 0; j < 8; ++j) {
            const int n = mt * 16 + j + hi * 8;
            out[(long)n * V + vox] = __expf(d[j]);
        }
    }
}

extern "C" void kernel_launch(void* const* d_in, const int* in_sizes, int n_in,
                              void* d_out, int out_size, void* d_ws, size_t ws_size,
                              hipStream_t stream) {
    const float* emb   = (const float*)d_in[0];  // [1,3,96,96,96]
    const float* cent  = (const float*)d_in[1];  // [1,64,3]
    const float* sigma = (const float*)d_in[2];  // [3]
    float* out = (float*)d_out;                  // [1,64,96,96,96]

    const long V = in_sizes[0] / 3;              // 884736 voxels
    // 8 waves/block, 16 voxels/wave -> 128 voxels per block.
    const long blocks = (V + 127) / 128;

    emb2prob_wmma_kernel<<<dim3((unsigned)blocks), dim3(256), 0, stream>>>(
        emb, cent, sigma, out, V);
}